// SSGC_10316511445630
// MI455X (gfx1250) — compile-verified
//
#include <hip/hip_runtime.h>

// ---------------------------------------------------------------------------
// SSGC on MI455X (gfx1250, wave32)
//
// Roofline: 8 SpMM rounds move ~13 GB through the cache hierarchy; the whole
// working set (51.2MB features x3 + 12.8MB edges) fits in the 192MB L2, so
// the SpMM phase is L2/atomic bound. FLOPs (~3 GFLOP) are negligible, so we
// keep full f32 precision and use V_WMMA_F32_16X16X4_F32 for the output GEMM.
// ---------------------------------------------------------------------------

#define NNODES    100000
#define INFEATS   128
#define NCLASSES  64
#define NEDGES    1600000
#define KITERS    8
#define ALPHA_F   0.05f

typedef __attribute__((ext_vector_type(2))) float v2f;
typedef __attribute__((ext_vector_type(8))) float v8f;

// Zero h and the first SpMM accumulator (float4-vectorized stores).
__global__ __launch_bounds__(256) void ssgc_zero2(float4* __restrict__ a,
                                                  float4* __restrict__ b,
                                                  int n4) {
    int i = blockIdx.x * blockDim.x + threadIdx.x;
    if (i < n4) {
        a[i] = make_float4(0.f, 0.f, 0.f, 0.f);
        b[i] = make_float4(0.f, 0.f, 0.f, 0.f);
    }
}

// One wave32 per edge: lane l handles feats [4l, 4l+4).
// acc[dst] += f[src]   via native global_atomic_add_f32.
__global__ __launch_bounds__(256) void ssgc_spmm(const float* __restrict__ f,
                                                 const int*   __restrict__ esrc,
                                                 const int*   __restrict__ edst,
                                                 float*       __restrict__ acc) {
    int gid  = blockIdx.x * blockDim.x + threadIdx.x;
    int lane = threadIdx.x & 31;
    // wave id is uniform across the wave -> readfirstlane makes the edge-index
    // loads scalar (s_load) instead of 32 replicated vector loads.
    int e = __builtin_amdgcn_readfirstlane(gid >> 5);   // 0 .. NEDGES-1 (exact grid)
    int s = esrc[e];
    int d = edst[e];
    const float4 v = *reinterpret_cast<const float4*>(f + (size_t)s * INFEATS + lane * 4);
    float* o = acc + (size_t)d * INFEATS + lane * 4;
    unsafeAtomicAdd(o + 0, v.x);
    unsafeAtomicAdd(o + 1, v.y);
    unsafeAtomicAdd(o + 2, v.z);
    unsafeAtomicAdd(o + 3, v.w);
}

// h = (h + (1-alpha)*fnext + alpha*feat0) / K   (exactly as the reference:
// the /K happens inside every iteration). Also zeroes the buffer that will
// be the next iteration's SpMM accumulator (fused to save a clear pass).
__global__ __launch_bounds__(256) void ssgc_blend(float* __restrict__ h,
                                                  const float* __restrict__ fnext,
                                                  const float* __restrict__ feat0,
                                                  float* __restrict__ zbuf) {
    int i = (blockIdx.x * blockDim.x + threadIdx.x) * 4;   // exact grid, no guard
    float4 hv = *reinterpret_cast<float4*>(h + i);
    float4 fv = *reinterpret_cast<const float4*>(fnext + i);
    float4 f0 = *reinterpret_cast<const float4*>(feat0 + i);
    const float c1 = 1.0f - ALPHA_F;
    const float c0 = ALPHA_F;
    const float ik = 1.0f / (float)KITERS;
    hv.x = (hv.x + c1 * fv.x + c0 * f0.x) * ik;
    hv.y = (hv.y + c1 * fv.y + c0 * f0.y) * ik;
    hv.z = (hv.z + c1 * fv.z + c0 * f0.z) * ik;
    hv.w = (hv.w + c1 * fv.w + c0 * f0.w) * ik;
    *reinterpret_cast<float4*>(h + i)    = hv;
    *reinterpret_cast<float4*>(zbuf + i) = make_float4(0.f, 0.f, 0.f, 0.f);
}

// out[N, 64] = h[N,128] @ W[64,128]^T + bias, via V_WMMA_F32_16X16X4_F32.
// One wave per 16x16 output tile; K=128 consumed in 32 steps of 4.
//
// Per ISA layouts (CDNA5 ISA):
//   A 16x4 f32 : lane L<16 holds A[M=L][K=0],A[M=L][K=1] in v[0],v[1];
//                lane L>=16 holds K=2,K=3                 -> offset hi*2
//   B 4x16 f32 : mirrored (row K striped across 16 lanes per VGPR)
//   C/D 16x16  : VGPR r -> M=r (lanes 0-15) / M=r+8 (lanes 16-31), N=lane&15
__global__ __launch_bounds__(256) void ssgc_gemm_wmma(const float* __restrict__ h,
                                                      const float* __restrict__ W,
                                                      const float* __restrict__ bias,
                                                      float* __restrict__ out) {
    int lane   = threadIdx.x & 31;
    int waveId = threadIdx.x >> 5;
    int tile   = blockIdx.x * 8 + waveId;   // 25000 tiles, grid exact
    int mt     = tile >> 2;                 // 0 .. 6249   (row tile)
    int nt     = tile & 3;                  // 0 .. 3      (class tile)
    int r16    = lane & 15;
    int hi     = lane >> 4;                 // 0 or 1

    const float* arow = h + (size_t)(mt * 16 + r16) * INFEATS + hi * 2;
    const float* brow = W + (size_t)(nt * 16 + r16) * INFEATS + hi * 2;

    v8f c = {};
#pragma unroll
    for (int kk = 0; kk < INFEATS / 4; ++kk) {
        v2f a = *reinterpret_cast<const v2f*>(arow + kk * 4);
        v2f b = *reinterpret_cast<const v2f*>(brow + kk * 4);
        // (neg_a, A, neg_b, B, c_mod, C, reuse_a, reuse_b)
        c = __builtin_amdgcn_wmma_f32_16x16x4_f32(false, a, false, b,
                                                  (short)0, c, false, false);
    }

    float bv = bias[nt * 16 + r16];
    float* obase = out + (size_t)(mt * 16 + hi * 8) * NCLASSES + nt * 16 + r16;
#pragma unroll
    for (int r = 0; r < 8; ++r) {
        obase[(size_t)r * NCLASSES] = c[r] + bv;
    }
}

extern "C" void kernel_launch(void* const* d_in, const int* in_sizes, int n_in,
                              void* d_out, int out_size, void* d_ws, size_t ws_size,
                              hipStream_t stream) {
    const float* feat = (const float*)d_in[0];   // [100000,128]
    const float* W    = (const float*)d_in[1];   // [64,128]
    const float* bias = (const float*)d_in[2];   // [64]
    const int*   esrc = (const int*)  d_in[3];   // [1.6M]
    const int*   edst = (const int*)  d_in[4];   // [1.6M]
    float*       out  = (float*)d_out;           // [100000,64]

    const size_t NF = (size_t)NNODES * INFEATS;  // 12.8M floats
    float* h    = (float*)d_ws;                  // 51.2 MB
    float* buf0 = h    + NF;                     // 51.2 MB
    float* buf1 = buf0 + NF;                     // 51.2 MB

    const int total4 = (int)(NF / 4);            // 3,200,000 (exact)

    // h = 0, first accumulator = 0
    ssgc_zero2<<<(total4 + 255) / 256, 256, 0, stream>>>(
        (float4*)h, (float4*)buf0, total4);

    const float* fin   = feat;   // SpMM input (never mutated)
    float*       acc   = buf0;   // SpMM output accumulator (pre-zeroed)
    float*       other = buf1;   // becomes the next accumulator (zeroed in blend)

    for (int k = 0; k < KITERS; ++k) {
        ssgc_spmm<<<NEDGES / 8, 256, 0, stream>>>(fin, esrc, edst, acc);
        ssgc_blend<<<total4 / 256, 256, 0, stream>>>(h, acc, feat, other);
        fin = acc;
        float* t = acc; acc = other; other = t;
    }

    // 25000 tiles (6250 x 4), 8 waves (tiles) per 256-thread block
    ssgc_gemm_wmma<<<(6250 * 4) / 8, 256, 0, stream>>>(h, W, bias, out);
}